// bboxing_46703474376913
// MI455X (gfx1250) — compile-verified
//
#include <hip/hip_runtime.h>
#include <hip/hip_bf16.h>

// ---------------------------------------------------------------------------
// Problem constants (reference: x [1,4,1024,1024] f32, MAX_BOXES=8192)
// ---------------------------------------------------------------------------
#define C_CH    4
#define H_DIM   1024
#define W_DIM   1024
#define S_PIX   (H_DIM * W_DIM)     // 1,048,576 pixels per channel
#define BIGL    S_PIX               // background label
#define NBOX    8192
#define TPB     256
#define NB_MEAN 256                 // partial-sum blocks per channel (4096 px each)
#define NBL_ALL ((C_CH * S_PIX) / TPB)   // 16384 blocks over C*S
#define NBL_CH  (S_PIX / TPB)            // 4096 blocks per channel

typedef float v2f __attribute__((ext_vector_type(2)));
typedef float v8f __attribute__((ext_vector_type(8)));

// --- optional CDNA5 async-load-to-LDS path (guarded; falls back cleanly) ---
#if defined(__has_builtin)
#  if __has_builtin(__builtin_amdgcn_global_load_async_to_lds_b32)
#    define HAVE_ASYNC_LDS 1
#  endif
#  if __has_builtin(__builtin_amdgcn_s_wait_asynccnt)
#    define HAVE_WAIT_ASYNC_BUILTIN 1
#  endif
#endif

__device__ __forceinline__ void wait_asynccnt0() {
#if defined(HAVE_WAIT_ASYNC_BUILTIN)
  __builtin_amdgcn_s_wait_asynccnt(0);
#else
  asm volatile("s_wait_asynccnt 0x0" ::: "memory");
#endif
}

// ---------------------------------------------------------------------------
// K1: per-channel partial sums via V_WMMA_F32_16X16X4_F32.
// Sum is permutation invariant, so the A-matrix lane mapping is irrelevant:
// each lane contributes 2 contiguous floats per WMMA; B = all-ones replicates
// every row-sum across 16 columns, so the grand total over D is 16x the true
// sum (divided out at the end). Fixed-order accumulation => deterministic.
// One block = 8 waves, 4096 contiguous floats of one channel.
// ---------------------------------------------------------------------------
__global__ __launch_bounds__(TPB) void k_mean_partial(
    const float* __restrict__ x, float* __restrict__ part) {
  const int blk  = blockIdx.x;            // c * NB_MEAN + b
  const int c    = blk / NB_MEAN;
  const int b    = blk - c * NB_MEAN;
  const int wave = threadIdx.x >> 5;
  const int lane = threadIdx.x & 31;
  const float* base = x + (size_t)c * S_PIX + (size_t)b * 4096 + (size_t)wave * 512;

  v8f acc = {};
  v2f ones; ones[0] = 1.0f; ones[1] = 1.0f;
#pragma unroll
  for (int it = 0; it < 8; ++it) {
    v2f a = *(const v2f*)(base + it * 64 + lane * 2);   // 8-byte aligned
    acc = __builtin_amdgcn_wmma_f32_16x16x4_f32(
        /*neg_a=*/false, a, /*neg_b=*/false, ones,
        /*c_mod=*/(short)0, acc, /*reuse_a=*/false, /*reuse_b=*/false);
  }
  float s = acc[0] + acc[1] + acc[2] + acc[3] + acc[4] + acc[5] + acc[6] + acc[7];
#pragma unroll
  for (int m = 16; m >= 1; m >>= 1) s += __shfl_xor(s, m, 32);

  __shared__ float wsum[8];
  if (lane == 0) wsum[wave] = s;
  __syncthreads();
  if (threadIdx.x == 0) {
    float t = 0.0f;
    for (int i = 0; i < 8; ++i) t += wsum[i];     // fixed order
    part[blk] = t * (1.0f / 16.0f);               // undo 16x column replication
  }
}

// K2: one block per channel folds the 256 partials (fixed-pairing tree).
__global__ __launch_bounds__(TPB) void k_mean_final(
    const float* __restrict__ part, float* __restrict__ means) {
  const int c = blockIdx.x;
  const int t = threadIdx.x;
  __shared__ float sm[NB_MEAN];
  sm[t] = part[c * NB_MEAN + t];
  __syncthreads();
  for (int off = NB_MEAN / 2; off > 0; off >>= 1) {
    if (t < off) sm[t] += sm[t + off];
    __syncthreads();
  }
  if (t == 0) means[c] = sm[0] * (1.0f / (float)S_PIX);
}

// ---------------------------------------------------------------------------
// K3: binarize + init union-find labels and component stats.
// ---------------------------------------------------------------------------
__global__ __launch_bounds__(TPB) void k_init(
    const float* __restrict__ x, const float* __restrict__ means,
    int* __restrict__ lab, int* __restrict__ sz,
    int* __restrict__ xmn, int* __restrict__ xmx, int* __restrict__ ymx) {
  const int i = blockIdx.x * TPB + threadIdx.x;          // over C*S
  if (i >= C_CH * S_PIX) return;
  __builtin_prefetch((const char*)x + (size_t)i * 4 + 16384, 0, 0); // global_prefetch_b8
  const int c = i / S_PIX;
  const int p = i - c * S_PIX;
  lab[i] = (x[i] >= means[c]) ? p : BIGL;
  sz[i]  = 0;
  xmn[i] = 0x7fffffff;
  xmx[i] = -1;
  ymx[i] = -1;
}

// ---------------------------------------------------------------------------
// K4: union-find merge over left/up edges (each 4-conn edge visited once).
// Min-union => final roots are the component-min indices (deterministic).
// ---------------------------------------------------------------------------
__device__ __forceinline__ int find_root(const int* __restrict__ L, int a) {
  int p = L[a];
  while (p != a) { a = p; p = L[a]; }
  return a;
}

__device__ __forceinline__ void merge_cc(int* L, int a, int b) {
  a = find_root(L, a);
  b = find_root(L, b);
  while (a != b) {
    if (a > b) { int t = a; a = b; b = t; }   // ensure a < b
    int old = atomicMin(&L[b], a);
    if (old == b) break;                      // b was root, now linked under a
    b = old;                                  // chase previous parent
  }
}

__global__ __launch_bounds__(TPB) void k_merge(int* __restrict__ lab) {
  const int i = blockIdx.x * TPB + threadIdx.x;
  if (i >= C_CH * S_PIX) return;
  const int c = i / S_PIX;
  const int p = i - c * S_PIX;
  int* L = lab + (size_t)c * S_PIX;
  if (L[p] == BIGL) return;                   // BIGL entries are immutable
  const int y  = p / W_DIM;
  const int xx = p - y * W_DIM;
  if (xx > 0 && L[p - 1]     != BIGL) merge_cc(L, p, p - 1);
  if (y  > 0 && L[p - W_DIM] != BIGL) merge_cc(L, p, p - W_DIM);
}

// K5: flatten every fg pixel to its root (concurrent path-compression safe).
__global__ __launch_bounds__(TPB) void k_flatten(int* __restrict__ lab) {
  const int i = blockIdx.x * TPB + threadIdx.x;
  if (i >= C_CH * S_PIX) return;
  const int c = i / S_PIX;
  const int p = i - c * S_PIX;
  int* L = lab + (size_t)c * S_PIX;
  if (L[p] != BIGL) L[p] = find_root(L, p);
}

// K6: per-component stats with integer atomics (order independent).
// ymin is implied by the root (min linear index): ymin = root / W.
__global__ __launch_bounds__(TPB) void k_stats(
    const int* __restrict__ lab, int* __restrict__ sz,
    int* __restrict__ xmn, int* __restrict__ xmx, int* __restrict__ ymx) {
  const int i = blockIdx.x * TPB + threadIdx.x;
  if (i >= C_CH * S_PIX) return;
  const int r = lab[i];
  if (r == BIGL) return;
  const int c  = i / S_PIX;
  const int p  = i - c * S_PIX;
  const int y  = p / W_DIM;
  const int xx = p - y * W_DIM;
  const int base = c * S_PIX + r;
  atomicAdd(&sz[base], 1);
  atomicMin(&xmn[base], xx);
  atomicMax(&xmx[base], xx);
  atomicMax(&ymx[base], y);
}

// ---------------------------------------------------------------------------
// Valid-root predicate shared by count/scatter phases.
// ---------------------------------------------------------------------------
__device__ __forceinline__ int valid_root(
    const int* __restrict__ lab, const int* __restrict__ sz,
    int c, int p, int ms, int Ms) {
  if (lab[(size_t)c * S_PIX + p] != p) return 0;   // fg root only
  const int s = sz[(size_t)c * S_PIX + p];
  return (s >= ms && s < Ms) ? 1 : 0;
}

__device__ __forceinline__ void load_thresholds(
    const float* __restrict__ min_p, const float* __restrict__ max_p,
    int* ms, int* Ms) {
  int a = (int)min_p[0]; if (a < 1) a = 1;          // int() truncation, >=1
  int b = (int)max_p[0]; if (b < a + 1) b = a + 1;
  *ms = a; *Ms = b;
}

// K7a: per-block count of valid roots (block = 256 contiguous pixels, never
// straddles a channel since S_PIX % TPB == 0).
__global__ __launch_bounds__(TPB) void k_count(
    const int* __restrict__ lab, const int* __restrict__ sz,
    const float* __restrict__ min_p, const float* __restrict__ max_p,
    int* __restrict__ blk_cnt) {
  const int bi = blockIdx.x;
  const int i  = bi * TPB + threadIdx.x;
  const int c  = i / S_PIX;
  const int p  = i - c * S_PIX;
  int ms, Ms; load_thresholds(min_p, max_p, &ms, &Ms);
  const int flag = valid_root(lab, sz, c, p, ms, Ms);

  __shared__ int sc[TPB];
  sc[threadIdx.x] = flag;
  __syncthreads();
  for (int off = TPB / 2; off > 0; off >>= 1) {
    if (threadIdx.x < off) sc[threadIdx.x] += sc[threadIdx.x + off];
    __syncthreads();
  }
  if (threadIdx.x == 0) blk_cnt[bi] = sc[0];
}

// K7b: per-channel exclusive scan of the 4096 block counts (one block/channel,
// 16 chunks of 256; chunks staged into LDS via the CDNA5 async path when the
// toolchain exposes it). Also records the per-channel total.
__global__ __launch_bounds__(TPB) void k_scan(
    const int* __restrict__ blk_cnt, int* __restrict__ blk_off,
    int* __restrict__ ch_total) {
  const int c = blockIdx.x;
  const int t = threadIdx.x;
  __shared__ int buf[TPB];
  __shared__ int sc[TPB];
  int base = 0;
  for (int chunk = 0; chunk < NBL_CH / TPB; ++chunk) {
    const int gi = c * NBL_CH + chunk * TPB + t;
#if defined(HAVE_ASYNC_LDS)
    // Builtin signature (from compiler diagnostic): generic int* params.
    __builtin_amdgcn_global_load_async_to_lds_b32(
        (int*)(blk_cnt + gi), (int*)(buf + t), /*offset=*/0, /*cpol=*/0);
    wait_asynccnt0();                        // this wave's copies are in LDS
    __syncthreads();                         // all waves' copies visible
#else
    buf[t] = blk_cnt[gi];
    __syncthreads();
#endif
    const int v = buf[t];
    // Hillis-Steele inclusive scan (deterministic order)
    sc[t] = v;
    __syncthreads();
    for (int off = 1; off < TPB; off <<= 1) {
      int u = (t >= off) ? sc[t - off] : 0;
      __syncthreads();
      sc[t] += u;
      __syncthreads();
    }
    blk_off[gi] = base + sc[t] - v;          // exclusive offset for block gi
    base += sc[TPB - 1];                     // uniform across the block
    __syncthreads();                         // protect sc/buf for next chunk
  }
  if (t == 0) ch_total[c] = base;
}

// K7c: grid-wide scatter — rank = channel-exclusive block offset + in-block
// scan rank; ascending root index per channel == raster order (top_k order).
__global__ __launch_bounds__(TPB) void k_scatter(
    const int* __restrict__ lab, const int* __restrict__ sz,
    const int* __restrict__ xmn, const int* __restrict__ xmx,
    const int* __restrict__ ymx,
    const float* __restrict__ min_p, const float* __restrict__ max_p,
    const int* __restrict__ blk_off,
    int* __restrict__ boxes, int* __restrict__ mask) {
  const int bi = blockIdx.x;
  const int t  = threadIdx.x;
  const int i  = bi * TPB + t;
  const int c  = i / S_PIX;
  const int p  = i - c * S_PIX;
  int ms, Ms; load_thresholds(min_p, max_p, &ms, &Ms);
  const int flag = valid_root(lab, sz, c, p, ms, Ms);

  __shared__ int sc[TPB];
  sc[t] = flag;
  __syncthreads();
  for (int off = 1; off < TPB; off <<= 1) {
    int u = (t >= off) ? sc[t - off] : 0;
    __syncthreads();
    sc[t] += u;
    __syncthreads();
  }
  const int rank = blk_off[bi] + sc[t] - flag;
  if (flag && rank < NBOX) {
    const int xm = xmn[(size_t)c * S_PIX + p];
    const int xM = xmx[(size_t)c * S_PIX + p];
    const int yM = ymx[(size_t)c * S_PIX + p];
    const int ym = p / W_DIM;                // root row = ymin
    int* bo = boxes + ((size_t)c * NBOX + rank) * 4;
    bo[0] = xm; bo[1] = ym; bo[2] = xM - xm + 1; bo[3] = yM - ym + 1;
    mask[(size_t)c * NBOX + rank] = 1;
  }
}

// K7d: zero-fill the unused tail of each channel's box list + mask.
__global__ __launch_bounds__(TPB) void k_fill(
    const int* __restrict__ ch_total,
    int* __restrict__ boxes, int* __restrict__ mask) {
  const int q = blockIdx.x * TPB + threadIdx.x;    // over C*NBOX
  if (q >= C_CH * NBOX) return;
  const int c    = q / NBOX;
  const int slot = q - c * NBOX;
  int tot = ch_total[c]; if (tot > NBOX) tot = NBOX;
  if (slot >= tot) {
    int* bo = boxes + (size_t)q * 4;
    bo[0] = 0; bo[1] = 0; bo[2] = 0; bo[3] = 0;
    mask[q] = 0;
  }
}

// ---------------------------------------------------------------------------
// Launch: all on `stream`, graph-capture safe (no malloc/sync/events).
// ws layout (bytes):
//   [0)          partial sums   C*NB_MEAN f32  (4 KB)
//   [8K)         means          C f32
//   [16K)        labels         C*S i32        (16 MB)
//   [+1*SB)      size           C*S i32
//   [+2*SB)      xmin           C*S i32
//   [+3*SB)      xmax           C*S i32
//   [+4*SB)      ymax           C*S i32
//   [+5*SB)      blk_cnt        NBL_ALL i32    (64 KB)
//   [+.. )       blk_off        NBL_ALL i32    (64 KB)
//   [+.. )       ch_total       C i32
// d_out: boxes C*NBOX*4 i32, then mask C*NBOX i32 (tuple concat in order).
// ---------------------------------------------------------------------------
extern "C" void kernel_launch(void* const* d_in, const int* in_sizes, int n_in,
                              void* d_out, int out_size, void* d_ws, size_t ws_size,
                              hipStream_t stream) {
  (void)in_sizes; (void)n_in; (void)out_size; (void)ws_size;

  const float* x     = (const float*)d_in[0];
  const float* min_p = (const float*)d_in[1];
  const float* max_p = (const float*)d_in[2];

  char* w = (char*)d_ws;
  const size_t SB = (size_t)C_CH * S_PIX * sizeof(int);   // 16 MB
  float* part   = (float*)(w);
  float* means  = (float*)(w + 8 * 1024);
  int*   lab    = (int*)  (w + 16 * 1024);
  int*   sz     = (int*)  (w + 16 * 1024 + 1 * SB);
  int*   xmn    = (int*)  (w + 16 * 1024 + 2 * SB);
  int*   xmx    = (int*)  (w + 16 * 1024 + 3 * SB);
  int*   ymx    = (int*)  (w + 16 * 1024 + 4 * SB);
  int*   bcnt   = (int*)  (w + 16 * 1024 + 5 * SB);
  int*   boff   = bcnt + NBL_ALL;
  int*   ctot   = boff + NBL_ALL;

  int* boxes = (int*)d_out;
  int* mask  = boxes + (size_t)C_CH * NBOX * 4;

  k_mean_partial<<<C_CH * NB_MEAN, TPB, 0, stream>>>(x, part);
  k_mean_final  <<<C_CH,           TPB, 0, stream>>>(part, means);
  k_init        <<<NBL_ALL,        TPB, 0, stream>>>(x, means, lab, sz, xmn, xmx, ymx);
  k_merge       <<<NBL_ALL,        TPB, 0, stream>>>(lab);
  k_flatten     <<<NBL_ALL,        TPB, 0, stream>>>(lab);
  k_stats       <<<NBL_ALL,        TPB, 0, stream>>>(lab, sz, xmn, xmx, ymx);
  k_count       <<<NBL_ALL,        TPB, 0, stream>>>(lab, sz, min_p, max_p, bcnt);
  k_scan        <<<C_CH,           TPB, 0, stream>>>(bcnt, boff, ctot);
  k_scatter     <<<NBL_ALL,        TPB, 0, stream>>>(lab, sz, xmn, xmx, ymx,
                                                     min_p, max_p, boff, boxes, mask);
  k_fill        <<<(C_CH * NBOX + TPB - 1) / TPB, TPB, 0, stream>>>(ctot, boxes, mask);
}